// WaveletEncoderLayer_72799695667501
// MI455X (gfx1250) — compile-verified
//
#include <hip/hip_runtime.h>
#include <math.h>

// Wavelet / weighted Lomb-Scargle encoder for MI455X (gfx1250).
// One wave32 per (batch, freq) pair; single fused sweep over 2048 samples
// using angle-sum identities (no second pass); TDM tensor_load_to_lds stages
// ts/ys rows into LDS; butterfly shuffle reduction across the wave.

#define NB   16
#define NF   512
#define NTS  2048
#define WAVES_PER_BLOCK 8
#define BLOCK 256

typedef __attribute__((ext_vector_type(4))) unsigned int u32x4;
typedef __attribute__((ext_vector_type(8))) int          i32x8;
typedef __attribute__((ext_vector_type(4))) int          i32x4;

// Issue one TDM descriptor: load a contiguous row of NTS f32 from global
// into LDS at byte offset lds_off.  D# layout per CDNA5 ISA ch.8.
__device__ __forceinline__ void tdm_load_row(const float* gsrc, unsigned int lds_off) {
    unsigned long long ga = (unsigned long long)(uintptr_t)gsrc;
    u32x4 g0;
    g0[0] = 1u;                                   // count=1, user descriptor
    g0[1] = lds_off;                              // lds_addr (bytes)
    g0[2] = (unsigned int)ga;                     // global_addr[31:0]
    g0[3] = (unsigned int)(ga >> 32) | 0x80000000u; // global_addr[56:32] | type=2
    i32x8 g1;
    g1[0] = (int)(2u << 16);                      // data_size = 2 (4 bytes)
    g1[1] = (int)(((unsigned)NTS & 0xFFFFu) << 16); // tensor_dim0[15:0] -> bits[63:48]
    g1[2] = (int)(1u << 16);                      // tensor_dim0 hi = 0, tensor_dim1 = 1
    g1[3] = (int)((unsigned)NTS << 16);           // tile_dim0 = 2048 -> bits[127:112]
    g1[4] = 1;                                    // tile_dim1 = 1, tile_dim2 = 0
    g1[5] = NTS;                                  // tensor_dim0_stride low 32
    g1[6] = 0;
    g1[7] = 0;
    i32x4 z4 = {0, 0, 0, 0};
#if defined(__clang_major__) && __clang_major__ >= 23
    i32x8 z8 = {0, 0, 0, 0, 0, 0, 0, 0};
    __builtin_amdgcn_tensor_load_to_lds(g0, g1, z4, z4, z8, 0);
#else
    __builtin_amdgcn_tensor_load_to_lds(g0, g1, z4, z4, 0);
#endif
}

__global__ __launch_bounds__(BLOCK)
void WaveletEncoderLayer_72799695667501_kernel(const float* __restrict__ ys,
                                               const float* __restrict__ ts,
                                               const float* __restrict__ freq,
                                               float* __restrict__ out) {
    __shared__ float s_ts[NTS];
    __shared__ float s_ys[NTS];

    const int b     = blockIdx.y;
    const int lane  = threadIdx.x & 31;
    const int wv    = threadIdx.x >> 5;
    const int f     = blockIdx.x * WAVES_PER_BLOCK + wv;

    // --- Stage ts[b,:] and ys[b,:] into LDS via Tensor Data Mover (wave 0) ---
    if (wv == 0) {
        tdm_load_row(ts + (size_t)b * NTS, (unsigned int)(uintptr_t)(&s_ts[0]));
        tdm_load_row(ys + (size_t)b * NTS, (unsigned int)(uintptr_t)(&s_ys[0]));
        __builtin_amdgcn_s_wait_tensorcnt(0);
    }
    __syncthreads();

    const float omega = 6.283185307179586f * freq[f];
    const float tau   = 0.5f * (s_ts[NTS / 2] + s_ts[NTS / 2 + 1]);
    const float Cc    = 0.012665147955292222f;   // 1/(8*pi^2)

    // Fused single sweep: 8 weighted sums.
    float acc[8];
#pragma unroll
    for (int i = 0; i < 8; ++i) acc[i] = 0.0f;

    for (int t = lane; t < NTS; t += 32) {
        float tt = s_ts[t];
        float y  = s_ys[t];
        float s, c;
        __sincosf(omega * tt, &s, &c);
        float dz = omega * (tt - tau);
        float w  = __expf(-Cc * dz * dz);
        float ws = w * s;
        float wc = w * c;
        acc[0] += w;                       // sum(w)
        acc[1] += ws;                      // sum(w*sin)
        acc[2] += wc;                      // sum(w*cos)
        acc[3]  = fmaf(ws, c, acc[3]);     // sum(w*sin*cos)
        acc[4]  = fmaf(ws, s, acc[4]);     // sum(w*sin^2)
        acc[5]  = fmaf(w,  y, acc[5]);     // sum(w*y)
        acc[6]  = fmaf(ws, y, acc[6]);     // sum(w*y*sin)
        acc[7]  = fmaf(wc, y, acc[7]);     // sum(w*y*cos)
    }

    // wave32 butterfly reduction (ds_swizzle path).
#pragma unroll
    for (int off = 16; off > 0; off >>= 1) {
#pragma unroll
        for (int i = 0; i < 8; ++i)
            acc[i] += __shfl_xor(acc[i], off, 32);
    }

    if (lane == 0) {
        const float sw = acc[0], sws = acc[1], swc = acc[2], swsc = acc[3],
                    swss = acc[4], swy = acc[5], swys = acc[6], swyc = acc[7];
        const float inv  = 1.0f / sw;
        const float sin1 = sws * inv;
        const float cos1 = swc * inv;
        const float sc   = swsc * inv;
        const float ss   = swss * inv;
        const float cc   = 1.0f - ss;      // (sum w*cos^2)/sum w = 1 - ss
        const float num  = 2.0f * (sc - sin1 * cos1);
        const float den  = (cc - cos1 * cos1) - (ss - sin1 * sin1);
        const float phi  = 0.5f * atan2f(num, den);   // = omega * time_shift

        float sphi, cphi;
        __sincosf(phi, &sphi, &cphi);
        // Rotate the pass-1 sums instead of a second sweep:
        // sin(theta - phi) = s*cphi - c*sphi ; cos(theta - phi) = c*cphi + s*sphi
        const float ys1 = swy * inv;
        const float ycs = (cphi * swyc + sphi * swys) * inv;  // <y*cos_shift>
        const float yss = (cphi * swys - sphi * swyc) * inv;  // <y*sin_shift>
        const float cs1 = (cphi * swc  + sphi * sws)  * inv;  // <cos_shift>
        const float ss1 = (cphi * sws  - sphi * swc)  * inv;  // <sin_shift>
        const float A = 2.0f * (ycs - ys1 * cs1);
        const float B = 2.0f * (yss - ys1 * ss1);

        const float psi = phi - omega * tau;   // omega*(time_shift - tau)
        float spsi, cpsi;
        __sincosf(psi, &spsi, &cpsi);
        const float a1 = cpsi * A - spsi * B;
        const float a2 = spsi * A + cpsi * B;
        const float wwp   = a1 * a1 + a2 * a2;
        const float phase = atan2f(a2, a1);

        const size_t idx = (size_t)b * NF + (size_t)f;
        out[0 * (size_t)NB * NF + idx] = wwp;
        out[1 * (size_t)NB * NF + idx] = phase;
        out[2 * (size_t)NB * NF + idx] = ys1;   // a0
        out[3 * (size_t)NB * NF + idx] = a1;
        out[4 * (size_t)NB * NF + idx] = a2;
    }
}

extern "C" void kernel_launch(void* const* d_in, const int* in_sizes, int n_in,
                              void* d_out, int out_size, void* d_ws, size_t ws_size,
                              hipStream_t stream) {
    (void)in_sizes; (void)n_in; (void)out_size; (void)d_ws; (void)ws_size;
    const float* ys   = (const float*)d_in[0];
    const float* ts   = (const float*)d_in[1];
    const float* freq = (const float*)d_in[2];
    float* out = (float*)d_out;

    dim3 block(BLOCK);
    dim3 grid(NF / WAVES_PER_BLOCK, NB);   // 64 x 16 blocks, one wave per (b,f)
    WaveletEncoderLayer_72799695667501_kernel<<<grid, block, 0, stream>>>(ys, ts, freq, out);
}